// Graph_41609643163904
// MI455X (gfx1250) — compile-verified
//
#include <hip/hip_runtime.h>
#include <cstdint>

// Problem constants (match reference)
#define BATCH  8192
#define SZ     48
#define DH     2
#define KK     25      // 5x5 weights per cell
#define NS     6       // rotating column slots in LDS window
#define WGSZ   32      // one wave32 per workgroup; lane == batch element

// TDM-friendly LDS layout: lds[slot][lane][row'], where the Tensor Data Mover
// writes each lane's 48-row column contiguously and inserts a 1-dword pad
// every 16 dwords (pad_interval=16, pad_amount=1).  Per-lane stride is
// 48 + 3 = 51 dwords (odd -> all 32 lanes hit distinct banks).
#define LSTRIDE 51
#define SLOTDW  (WGSZ * LSTRIDE)   // 1632 dwords per slot; 6 slots = 39,168 B

typedef uint32_t u32x4 __attribute__((ext_vector_type(4)));
typedef uint32_t u32x8 __attribute__((ext_vector_type(8)));

// Padded in-lane dword offset of row y (TDM pads 1 dword after every 16).
__device__ __forceinline__ constexpr int lofs(int row) { return row + (row >> 4); }

// One TDM descriptor: 3D tile (X=1 elem, Y=48 rows stride 48, Z=32 lanes
// stride 2304) -> LDS contiguous with padding.  All fields wave-uniform.
__device__ __forceinline__ void tdm_load_col(uint64_t gaddr, uint32_t lds_byte)
{
    u32x4 g0 = { 1u,                                   // count=1, user desc
                 lds_byte,                             // lds_addr (bytes)
                 (uint32_t)gaddr,                      // global_addr[31:0]
                 (uint32_t)(gaddr >> 32) | 0x80000000u /* addr[56:32] | type=2 */ };
    u32x8 g1 = { (2u << 16) | (1u << 20) | (3u << 22), // data_size=4B, pad_en,
                                                       // pad every 16dw, +1dw
                 (1u  << 16),                          // tensor_dim0 = 1
                 (48u << 16),                          // tensor_dim1 = 48
                 (1u  << 16),                          // tile_dim0  = 1
                 48u | (32u << 16),                    // tile_dim1=48, tile_dim2=32
                 48u,                                  // tensor_dim0_stride = 48
                 (2304u << 16),                        // tensor_dim1_stride = 2304
                 0u };
    u32x4 g2 = { 32u, 0u, 0u, 0u };                    // tensor_dim2 = 32
    u32x4 g3 = { 0u, 0u, 0u, 0u };
    asm volatile("tensor_load_to_lds %0, %1, %2, %3"
                 :: "s"(g0), "s"(g1), "s"(g2), "s"(g3) : "memory");
}

__global__ __launch_bounds__(WGSZ) void seq_stencil_sweep(
    const float* __restrict__ inp,
    const float* __restrict__ wgt,
    float* __restrict__ out)
{
    __shared__ float lds[NS * SLOTDW];

    const int lane       = threadIdx.x;
    const int laneoff    = lane * LSTRIDE;
    const uint32_t lds0  = (uint32_t)(uintptr_t)(void*)&lds[0];
    const uint32_t gbase = (uint32_t)(blockIdx.x * WGSZ + lane) * (SZ * SZ);
    const uint64_t outB  = (uint64_t)(uintptr_t)out;
    // Wave-uniform global byte address of (batch b0, row 0, col 0)
    const uint64_t colBase = (uint64_t)(uintptr_t)inp
                           + (uint64_t)(blockIdx.x * WGSZ) * (SZ * SZ * 4);

    // ---- TDM preload: columns 0..4 -> slots 0..4 (slot = column % 6) ----
    #pragma unroll
    for (int c = 0; c < 5; ++c)
        tdm_load_col(colBase + (uint64_t)c * 4u,
                     lds0 + (uint32_t)(c * SLOTDW) * 4u);

    float r[5][5];          // register window: r[j][row%5], column (x-2+j)
    int s0 = 0;             // LDS slot holding column (x-2)

    for (int x = DH; x < SZ - DH; ++x) {
        __builtin_amdgcn_s_wait_tensorcnt(0);            // col x+2 landed in LDS
        asm volatile("s_wait_asynccnt 0" ::: "memory");  // prior column store drained

        int s1 = s0 + 1; if (s1 >= NS) s1 -= NS;
        int s2 = s1 + 1; if (s2 >= NS) s2 -= NS;
        int s3 = s2 + 1; if (s3 >= NS) s3 -= NS;
        int s4 = s3 + 1; if (s4 >= NS) s4 -= NS;
        int sl = s0 - 1; if (sl < 0)   sl += NS;         // slot for column x+3

        // One TDM op fetches the whole wave's next column; overlaps compute.
        if (x + 3 < SZ)
            tdm_load_col(colBase + (uint64_t)(x + 3) * 4u,
                         lds0 + (uint32_t)(sl * SLOTDW) * 4u);

        const int base[5] = { s0 * SLOTDW + laneoff, s1 * SLOTDW + laneoff,
                              s2 * SLOTDW + laneoff, s3 * SLOTDW + laneoff,
                              s4 * SLOTDW + laneoff };

        // Prologue: rows 0..4 of the 5 active columns -> registers
        #pragma unroll
        for (int j = 0; j < 5; ++j)
            #pragma unroll
            for (int i = 0; i < 5; ++i)
                r[j][i] = lds[base[j] + lofs(i)];

        const float* wx = wgt + x * KK;
        // Fully unrolled sweep: row%5 register indices are compile-time, so
        // the window "shift" is pure renaming (no v_mov), and the TDM pad
        // correction lofs() folds to constants.
        #pragma unroll
        for (int y = DH; y < SZ - DH; ++y) {
            const float* w = wx + y * (SZ * KK);   // uniform -> s_load
            const int i0 = (y - 2) % 5, i1 = (y - 1) % 5, i2 = y % 5,
                      i3 = (y + 1) % 5, i4 = (y + 2) % 5;
            float v = r[2][i2];
            // center column: center row overwritten, others accumulate
            r[2][i0] += v * w[0 * 5 + 2];
            r[2][i1] += v * w[1 * 5 + 2];
            r[2][i2]  = v * w[2 * 5 + 2];
            r[2][i3] += v * w[3 * 5 + 2];
            r[2][i4] += v * w[4 * 5 + 2];
            #pragma unroll
            for (int jj = 0; jj < 4; ++jj) {
                const int j = (jj < 2) ? jj : jj + 1;   // j in {0,1,3,4}
                r[j][i0] += v * w[0 * 5 + j];
                r[j][i1] += v * w[1 * 5 + j];
                r[j][i2] += v * w[2 * 5 + j];
                r[j][i3] += v * w[3 * 5 + j];
                r[j][i4] += v * w[4 * 5 + j];
            }
            // Flush finalized row y-2; reuse its register slot for row y+3
            #pragma unroll
            for (int j = 0; j < 5; ++j) {
                lds[base[j] + lofs(y - DH)] = r[j][i0];
                if (y + 3 < SZ)
                    r[j][i0] = lds[base[j] + lofs(y + 3)];
            }
        }
        // Epilogue: rows 44..47 still live in registers
        #pragma unroll
        for (int i = 0; i < 4; ++i) {
            const int row = SZ - 4 + i;
            #pragma unroll
            for (int j = 0; j < 5; ++j)
                lds[base[j] + lofs(row)] = r[j][row % 5];
        }

        // DS writes must land before the async store reads LDS
        // (async ops are unordered vs DS ops).
        asm volatile("s_wait_dscnt 0" ::: "memory");

        // Async DMA finalized column x-2 (slot s0) LDS -> global
        #pragma unroll
        for (int y = 0; y < SZ; ++y) {
            uint32_t loff = lds0 + (uint32_t)((s0 * SLOTDW + laneoff + lofs(y)) * 4);
            uint32_t goff = (gbase + (uint32_t)(y * SZ + (x - DH))) * 4u;
            asm volatile("global_store_async_from_lds_b32 %0, %1, %2"
                         :: "v"(goff), "v"(loff), "s"(outB) : "memory");
        }
        s0 = s1;
    }

    // Columns 44..47 (slots s0..s0+3) are final; DMA them out.
    asm volatile("s_wait_asynccnt 0" ::: "memory");
    #pragma unroll
    for (int cc = 0; cc < 4; ++cc) {
        int cs = s0 + cc; if (cs >= NS) cs -= NS;
        #pragma unroll
        for (int y = 0; y < SZ; ++y) {
            uint32_t loff = lds0 + (uint32_t)((cs * SLOTDW + laneoff + lofs(y)) * 4);
            uint32_t goff = (gbase + (uint32_t)(y * SZ + (SZ - 4 + cc))) * 4u;
            asm volatile("global_store_async_from_lds_b32 %0, %1, %2"
                         :: "v"(goff), "v"(loff), "s"(outB) : "memory");
        }
    }
    // s_endpgm performs an implicit wait-idle (flushes ASYNCcnt/TENSORcnt).
}

extern "C" void kernel_launch(void* const* d_in, const int* in_sizes, int n_in,
                              void* d_out, int out_size, void* d_ws, size_t ws_size,
                              hipStream_t stream) {
    (void)in_sizes; (void)n_in; (void)ws_size; (void)d_ws; (void)out_size;
    const float* inp = (const float*)d_in[0];   // (8192, 2304) f32
    const float* wgt = (const float*)d_in[1];   // (48, 48, 5, 5) f32
    float* out = (float*)d_out;                 // (8192, 2304) f32
    seq_stencil_sweep<<<BATCH / WGSZ, WGSZ, 0, stream>>>(inp, wgt, out);
}